// ScaledDotProductAttention_6932077216063
// MI455X (gfx1250) — compile-verified
//
#include <hip/hip_runtime.h>
#include <math.h>

// CDNA5 / gfx1250 flash-style attention (fp32, V_WMMA_F32_16X16X4_F32).
// B=2, H=12, S=2048, D=64. Outputs: [context | attn_prob] concatenated.

typedef __attribute__((ext_vector_type(2))) float v2f;
typedef __attribute__((ext_vector_type(8))) float v8f;

#define ATT_B 2
#define ATT_H 12
#define ATT_S 2048
#define ATT_D 64
#define ATT_SCALE 0.125f       // 1/sqrt(64)
#define ATT_MASKVAL -1.0e9f

// ---------------------------------------------------------------------------
// Kernel 1: scores = scale * Q K^T, mask, softmax, write attn_prob.
// Grid: x = S/16 (query row tile), y = B*H. Block: 256 threads (8 waves).
// Wave w covers key columns [w*256, w*256+256) as 16 WMMA tiles of 16x16.
// ---------------------------------------------------------------------------
__global__ void __launch_bounds__(256, 1)
attn_qk_softmax_kernel(const float* __restrict__ Q,
                       const float* __restrict__ K,
                       const unsigned char* __restrict__ mask, // [B,1,S,S] bool
                       float* __restrict__ probs)              // [B,H,S,S]
{
    const int qt   = blockIdx.x;           // 0..127
    const int bh   = blockIdx.y;           // 0..23
    const int b    = bh / ATT_H;
    const int lane = threadIdx.x & 31;
    const int wave = threadIdx.x >> 5;     // 0..7
    const int half = lane >> 4;            // 0/1 (lane group)
    const int lidx = lane & 15;
    const int q0   = qt * 16;

    const float* Qb = Q + (size_t)bh * ATT_S * ATT_D;
    const float* Kb = K + (size_t)bh * ATT_S * ATT_D;
    const unsigned char* Mb = mask + (size_t)b * ATT_S * ATT_S;
    float* Pb = probs + (size_t)bh * ATT_S * ATT_S;

    __shared__ float red[8][16];
    __shared__ float rowstat[16];

    // ---- A fragments (Q tile 16x64): 16 k-steps of 16x4 f32.
    // ISA layout: lane M = lidx; VGPR0/1 hold K = 2*half + {0,1}.
    v2f afrag[16];
#pragma unroll
    for (int kk = 0; kk < 16; ++kk) {
        const float* p = Qb + (size_t)(q0 + lidx) * ATT_D + kk * 4 + half * 2;
        afrag[kk] = *(const v2f*)p;
    }

    // ---- Accumulate 16 column tiles: scores[16 rows][wave's 256 cols].
    v8f acc[16];
#pragma unroll
    for (int t = 0; t < 16; ++t) acc[t] = (v8f){};

#pragma unroll
    for (int kk = 0; kk < 16; ++kk) {
#pragma unroll
        for (int t = 0; t < 16; ++t) {
            const int n0 = wave * 256 + t * 16;
            // B (K^T) 4x16 tile: lane N = lidx; VGPR0/1 hold K-rows 2*half+{0,1}.
            // B[k][n] = K[n0+n][kk*4+k]  -> contiguous float2 per lane.
            const float* p = Kb + (size_t)(n0 + lidx) * ATT_D + kk * 4 + half * 2;
            v2f bfrag = *(const v2f*)p;
            acc[t] = __builtin_amdgcn_wmma_f32_16x16x4_f32(
                false, afrag[kk], false, bfrag, (short)0, acc[t], false, false);
        }
    }

    // ---- Scale + mask + per-lane row max.
    // C/D layout: lane holds col N = n0+lidx; VGPR i holds row M = i + 8*half.
    const int qrow = q0 + half * 8;
    float pmax[8];
#pragma unroll
    for (int i = 0; i < 8; ++i) pmax[i] = -INFINITY;

#pragma unroll
    for (int t = 0; t < 16; ++t) {
        const int col = wave * 256 + t * 16 + lidx;
#pragma unroll
        for (int i = 0; i < 8; ++i) {
            float v = acc[t][i] * ATT_SCALE;
            if (Mb[(size_t)(qrow + i) * ATT_S + col]) v = ATT_MASKVAL;
            acc[t][i] = v;
            pmax[i] = fmaxf(pmax[i], v);
        }
    }

    // ---- Row max: reduce across the 16 lanes holding a row, then across waves.
#pragma unroll
    for (int i = 0; i < 8; ++i) {
        float m = pmax[i];
        m = fmaxf(m, __shfl_xor(m, 1, 16));
        m = fmaxf(m, __shfl_xor(m, 2, 16));
        m = fmaxf(m, __shfl_xor(m, 4, 16));
        m = fmaxf(m, __shfl_xor(m, 8, 16));
        pmax[i] = m;
    }
    if (lidx == 0) {
#pragma unroll
        for (int i = 0; i < 8; ++i) red[wave][half * 8 + i] = pmax[i];
    }
    __syncthreads();
    if (threadIdx.x < 16) {
        float m = red[0][threadIdx.x];
#pragma unroll
        for (int w = 1; w < 8; ++w) m = fmaxf(m, red[w][threadIdx.x]);
        rowstat[threadIdx.x] = m;
    }
    __syncthreads();

    float rmax[8];
#pragma unroll
    for (int i = 0; i < 8; ++i) rmax[i] = rowstat[half * 8 + i];

    // ---- exp and row-sum.
    float psum[8];
#pragma unroll
    for (int i = 0; i < 8; ++i) psum[i] = 0.0f;
#pragma unroll
    for (int t = 0; t < 16; ++t) {
#pragma unroll
        for (int i = 0; i < 8; ++i) {
            float e = __expf(acc[t][i] - rmax[i]);
            acc[t][i] = e;
            psum[i] += e;
        }
    }
#pragma unroll
    for (int i = 0; i < 8; ++i) {
        float s = psum[i];
        s += __shfl_xor(s, 1, 16);
        s += __shfl_xor(s, 2, 16);
        s += __shfl_xor(s, 4, 16);
        s += __shfl_xor(s, 8, 16);
        psum[i] = s;
    }
    __syncthreads();                  // everyone done reading rowstat/red
    if (lidx == 0) {
#pragma unroll
        for (int i = 0; i < 8; ++i) red[wave][half * 8 + i] = psum[i];
    }
    __syncthreads();
    if (threadIdx.x < 16) {
        float s = red[0][threadIdx.x];
#pragma unroll
        for (int w = 1; w < 8; ++w) s += red[w][threadIdx.x];
        rowstat[threadIdx.x] = s;
    }
    __syncthreads();

    float rinv[8];
#pragma unroll
    for (int i = 0; i < 8; ++i) rinv[i] = 1.0f / rowstat[half * 8 + i];

    // ---- Write probabilities (16 lanes = 64B contiguous per half-group).
#pragma unroll
    for (int t = 0; t < 16; ++t) {
        const int col = wave * 256 + t * 16 + lidx;
#pragma unroll
        for (int i = 0; i < 8; ++i) {
            Pb[(size_t)(qrow + i) * ATT_S + col] = acc[t][i] * rinv[i];
        }
    }
}

// ---------------------------------------------------------------------------
// Kernel 2: context = attn_prob @ V.
// Grid: x = S/32 (32 query rows per block), y = B*H. Block: 256 threads.
// Each wave owns one 16x16 output tile; K=2048 in steps of 4 with 4
// independent accumulators to break the WMMA dependency chain.
// ---------------------------------------------------------------------------
__global__ void __launch_bounds__(256, 1)
attn_pv_kernel(const float* __restrict__ probs,   // [B,H,S,S]
               const float* __restrict__ V,       // [B,H,S,D]
               float* __restrict__ ctx)           // [B,H,S,D]
{
    const int qt   = blockIdx.x;            // 0..63
    const int bh   = blockIdx.y;            // 0..23
    const int lane = threadIdx.x & 31;
    const int wave = threadIdx.x >> 5;
    const int half = lane >> 4;
    const int lidx = lane & 15;

    const int r  = wave >> 2;               // row tile within block (0..1)
    const int c  = wave & 3;                // col tile (0..3)
    const int q0 = qt * 32 + r * 16;
    const int d0 = c * 16;

    const float* Pb = probs + (size_t)bh * ATT_S * ATT_S;
    const float* Vb = V + (size_t)bh * ATT_S * ATT_D;

    v8f acc0 = (v8f){}, acc1 = (v8f){}, acc2 = (v8f){}, acc3 = (v8f){};

#pragma unroll 4
    for (int k = 0; k < ATT_S; k += 16) {
        // 4 k-steps per iteration, one per accumulator -> 4-way ILP.
#pragma unroll
        for (int j = 0; j < 4; ++j) {
            const int kb = k + j * 4;
            // A (P) 16x4: lane M = lidx, K = kb + 2*half + {0,1}: float2.
            const float* ap = Pb + (size_t)(q0 + lidx) * ATT_S + kb + half * 2;
            v2f a = *(const v2f*)ap;
            // B (V) 4x16: lane N = lidx (d-col), K-rows = kb + 2*half + {0,1}.
            const float* bp = Vb + (size_t)(kb + half * 2) * ATT_D + d0 + lidx;
            v2f bfrag;
            bfrag.x = bp[0];
            bfrag.y = bp[ATT_D];
            if (j == 0)
                acc0 = __builtin_amdgcn_wmma_f32_16x16x4_f32(false, a, false, bfrag, (short)0, acc0, false, false);
            else if (j == 1)
                acc1 = __builtin_amdgcn_wmma_f32_16x16x4_f32(false, a, false, bfrag, (short)0, acc1, false, false);
            else if (j == 2)
                acc2 = __builtin_amdgcn_wmma_f32_16x16x4_f32(false, a, false, bfrag, (short)0, acc2, false, false);
            else
                acc3 = __builtin_amdgcn_wmma_f32_16x16x4_f32(false, a, false, bfrag, (short)0, acc3, false, false);
        }
    }

    v8f total = (acc0 + acc1) + (acc2 + acc3);

    const int qrow = q0 + half * 8;
#pragma unroll
    for (int i = 0; i < 8; ++i) {
        ctx[((size_t)bh * ATT_S + qrow + i) * ATT_D + d0 + lidx] = total[i];
    }
}

// ---------------------------------------------------------------------------
extern "C" void kernel_launch(void* const* d_in, const int* in_sizes, int n_in,
                              void* d_out, int out_size, void* d_ws, size_t ws_size,
                              hipStream_t stream) {
    (void)in_sizes; (void)n_in; (void)out_size; (void)d_ws; (void)ws_size;

    const float* Q = (const float*)d_in[0];
    const float* K = (const float*)d_in[1];
    const float* V = (const float*)d_in[2];
    const unsigned char* mask = (const unsigned char*)d_in[3]; // jnp bool: 1B/elem

    float* out   = (float*)d_out;
    float* ctx   = out;                                              // [B,H,S,D]
    float* probs = out + (size_t)ATT_B * ATT_H * ATT_S * ATT_D;      // [B,H,S,S]

    dim3 block(256, 1, 1);
    dim3 grid1(ATT_S / 16, ATT_B * ATT_H, 1);   // 128 x 24
    attn_qk_softmax_kernel<<<grid1, block, 0, stream>>>(Q, K, mask, probs);

    dim3 grid2(ATT_S / 32, ATT_B * ATT_H, 1);   // 64 x 24
    attn_pv_kernel<<<grid2, block, 0, stream>>>(probs, V, ctx);
}